// MultiBoxLoss_18004502904844
// MI455X (gfx1250) — compile-verified
//
#include <hip/hip_runtime.h>
#include <math.h>

// MultiBoxLoss (RetinaFace/SSD) for MI455X gfx1250.
// B=64 images, P=16800 priors, O=64 truths, 2 classes.
// Bandwidth-bound (~69MB traffic, ~1-2 GFLOP) -> single streaming pass per
// stage, truths staged in LDS, top-k via radix select (no sort), block-level
// sums reduced through the matrix pipe (v_wmma_f32_16x16x4_f32).

static constexpr int B_ = 64;
static constexpr int P_ = 16800;
static constexpr int O_ = 64;
static constexpr float THRESH = 0.35f;
static constexpr int NEGPOS = 7;
static constexpr float VAR0 = 0.1f;
static constexpr float VAR1 = 0.2f;
static constexpr int PBLK = 66; // ceil(16800/256)

typedef float v2f __attribute__((ext_vector_type(2)));
typedef float v8f __attribute__((ext_vector_type(8)));

// ---- wave32 sum reduction via the WMMA pipe -------------------------------
// A (16x4 f32): lanes 0-15 give rows via VGPR0=K0 / VGPR1=K1, lanes 16-31 via
// VGPR0=K2 / VGPR1=K3. With a={v,0} and B=ones: D[m,n] = v[m] + v[m+16].
// C/D layout: vgpr j, lanes 0-15 -> M=j, lanes 16-31 -> M=8+j. Summing the 8
// accumulator VGPRs gives half-wave sums; one xor-16 shuffle completes it.
__device__ __forceinline__ float wave_sum32(float v) {
  v2f a; a[0] = v; a[1] = 0.0f;
  v2f b; b[0] = 1.0f; b[1] = 1.0f;
  v8f c = {0.f, 0.f, 0.f, 0.f, 0.f, 0.f, 0.f, 0.f};
  c = __builtin_amdgcn_wmma_f32_16x16x4_f32(false, a, false, b, (short)0, c,
                                            false, false);
  float s = c[0] + c[1] + c[2] + c[3] + c[4] + c[5] + c[6] + c[7];
  s += __shfl_xor(s, 16, 32);
  return s; // full wave sum in every lane
}

// Block sum (256 threads = 8 waves). Valid on thread 0. All threads must call.
__device__ __forceinline__ float block_sum(float v, float* lds8) {
  float w = wave_sum32(v);
  int wave = threadIdx.x >> 5;
  if ((threadIdx.x & 31) == 0) lds8[wave] = w;
  __syncthreads();
  float tot = 0.f;
  if (threadIdx.x == 0) {
    for (int i = 0; i < 8; ++i) tot += lds8[i];
  }
  __syncthreads();
  return tot;
}

__device__ __forceinline__ float sl1(float x) {
  float a = fabsf(x);
  return a < 1.f ? 0.5f * a * a : a - 0.5f;
}

// ---------------------------------------------------------------------------
__global__ void k_init(float* acc, int* numpos, int* numposlm) {
  int i = threadIdx.x;
  if (i < 4) acc[i] = 0.f;
  if (i < B_) { numpos[i] = 0; numposlm[i] = 0; }
}

// Per-prior best truth: grid (PBLK, B) x 256
__global__ void k_match(const float* __restrict__ priors,
                        const float* __restrict__ targets,
                        float* __restrict__ bto, int* __restrict__ bti) {
  __shared__ float4 tr[O_];
  int b = blockIdx.y;
  for (int t = threadIdx.x; t < O_; t += blockDim.x) {
    const float* row = targets + ((size_t)b * O_ + t) * 15;
    tr[t] = make_float4(row[0], row[1], row[2], row[3]);
  }
  __syncthreads();
  int p = blockIdx.x * blockDim.x + threadIdx.x;
  if (p >= P_) return;
  float4 pr = *(const float4*)(priors + (size_t)p * 4); // cx,cy,w,h
  float x1 = pr.x - 0.5f * pr.z, y1 = pr.y - 0.5f * pr.w;
  float x2 = pr.x + 0.5f * pr.z, y2 = pr.y + 0.5f * pr.w;
  float ap = (x2 - x1) * (y2 - y1);
  float best = -1.f; int bidx = 0;
  for (int t = 0; t < O_; ++t) {
    float4 tb = tr[t];
    float w = fminf(tb.z, x2) - fmaxf(tb.x, x1);
    float h = fminf(tb.w, y2) - fmaxf(tb.y, y1);
    float inter = fmaxf(w, 0.f) * fmaxf(h, 0.f);
    float at = (tb.z - tb.x) * (tb.w - tb.y);
    float o = inter / (at + ap - inter);
    if (o > best) { best = o; bidx = t; }
  }
  size_t idx = (size_t)b * P_ + p;
  bto[idx] = best;
  bti[idx] = bidx;
}

// Per-truth best prior: grid (O, B) x 256, deterministic block reduction.
__global__ void k_bestprior(const float* __restrict__ priors,
                            const float* __restrict__ targets,
                            float* __restrict__ bpo, int* __restrict__ bpi) {
  __shared__ unsigned long long red[256];
  int t = blockIdx.x, b = blockIdx.y;
  const float* row = targets + ((size_t)b * O_ + t) * 15;
  float tx1 = row[0], ty1 = row[1], tx2 = row[2], ty2 = row[3];
  float at = (tx2 - tx1) * (ty2 - ty1);
  unsigned long long best = 0ull;
  for (int p = threadIdx.x; p < P_; p += blockDim.x) {
    float4 pr = *(const float4*)(priors + (size_t)p * 4);
    float x1 = pr.x - 0.5f * pr.z, y1 = pr.y - 0.5f * pr.w;
    float x2 = pr.x + 0.5f * pr.z, y2 = pr.y + 0.5f * pr.w;
    float w = fminf(tx2, x2) - fmaxf(tx1, x1);
    float h = fminf(ty2, y2) - fmaxf(ty1, y1);
    float inter = fmaxf(w, 0.f) * fmaxf(h, 0.f);
    float ap = (x2 - x1) * (y2 - y1);
    float o = fmaxf(inter / (at + ap - inter), 0.f);
    // pack (overlap, ~p): ties pick lowest prior index (matches stable argmax)
    unsigned long long key =
        ((unsigned long long)__float_as_uint(o) << 32) | (unsigned)(~(unsigned)p);
    best = (key > best) ? key : best;
  }
  red[threadIdx.x] = best;
  __syncthreads();
  for (int s = 128; s > 0; s >>= 1) {
    if (threadIdx.x < s) {
      unsigned long long o2 = red[threadIdx.x + s];
      if (o2 > red[threadIdx.x]) red[threadIdx.x] = o2;
    }
    __syncthreads();
  }
  if (threadIdx.x == 0) {
    bpo[b * O_ + t] = __uint_as_float((unsigned)(red[0] >> 32));
    bpi[b * O_ + t] = (int)~(unsigned)red[0];
  }
}

// Scatter overrides with reference scatter semantics (reads pre-scatter vals,
// duplicate indices: last write wins). grid B x 64.
__global__ void k_override(const float* __restrict__ bpo,
                           const int* __restrict__ bpi,
                           float* __restrict__ bto, int* __restrict__ bti) {
  __shared__ float orig[O_], vo[O_];
  __shared__ int pidx[O_];
  int b = blockIdx.x;
  int t = threadIdx.x;
  if (t < O_) {
    pidx[t] = bpi[b * O_ + t];
    vo[t] = bpo[b * O_ + t];
    orig[t] = bto[(size_t)b * P_ + pidx[t]];
  }
  __syncthreads();
  if (t == 0) {
    for (int i = 0; i < O_; ++i) {
      size_t idx = (size_t)b * P_ + pidx[i];
      bti[idx] = i;                                   // unconditional in ref
      bto[idx] = (vo[i] >= 0.2f) ? 2.0f : orig[i];    // where(valid, 2, old)
    }
  }
}

// Encode + smooth-L1 partials + conf loss. grid (PBLK, B) x 256.
__global__ void k_loss(const float* __restrict__ loc_data,
                       const float* __restrict__ conf_data,
                       const float* __restrict__ landm_data,
                       const float* __restrict__ priors,
                       const float* __restrict__ targets,
                       const float* __restrict__ bto,
                       const int* __restrict__ bti,
                       float* __restrict__ lossc, unsigned char* __restrict__ flags,
                       int* __restrict__ numpos, int* __restrict__ numposlm,
                       float* __restrict__ acc) {
  __shared__ float str[O_ * 16]; // targets rows, padded 15->16
  __shared__ float lsum[8];
  __shared__ int cnt[2];
  int b = blockIdx.y;
  for (int i = threadIdx.x; i < O_ * 15; i += blockDim.x)
    str[(i / 15) * 16 + (i % 15)] = targets[(size_t)b * O_ * 15 + i];
  if (threadIdx.x < 2) cnt[threadIdx.x] = 0;
  __syncthreads();

  int p = blockIdx.x * blockDim.x + threadIdx.x;
  bool act = p < P_;
  float ll = 0.f, llm = 0.f;
  if (act) {
    size_t idx = (size_t)b * P_ + p;
    float o = bto[idx];
    int t = bti[idx];
    const float* trow = &str[t * 16];
    float lab = trow[14];
    float conf = (o < THRESH) ? 0.f : lab;
    bool pos = conf != 0.f;
    bool pos1 = conf > 0.f;

    float4 pr = *(const float4*)(priors + (size_t)p * 4);
    float ivx = 1.f / (VAR0 * pr.z), ivy = 1.f / (VAR0 * pr.w);

    if (pos) {
      float gcx = ((trow[0] + trow[2]) * 0.5f - pr.x) * ivx;
      float gcy = ((trow[1] + trow[3]) * 0.5f - pr.y) * ivy;
      float gw = logf((trow[2] - trow[0]) / pr.z) * (1.f / VAR1);
      float gh = logf((trow[3] - trow[1]) / pr.w) * (1.f / VAR1);
      const float* ld = loc_data + idx * 4;
      ll = sl1(ld[0] - gcx) + sl1(ld[1] - gcy) + sl1(ld[2] - gw) + sl1(ld[3] - gh);
      atomicAdd(&cnt[0], 1);
    }
    if (pos1) {
      const float* lmd = landm_data + idx * 10;
#pragma unroll
      for (int i = 0; i < 5; ++i) {
        float tx = (trow[4 + 2 * i] - pr.x) * ivx;
        float ty = (trow[5 + 2 * i] - pr.y) * ivy;
        llm += sl1(lmd[2 * i] - tx) + sl1(lmd[2 * i + 1] - ty);
      }
      atomicAdd(&cnt[1], 1);
    }
    // conf loss: logsumexp - gathered (== the CE term used for mining AND sum)
    const float* cd = conf_data + idx * 2;
    float c0 = cd[0], c1 = cd[1];
    float m = fmaxf(c0, c1);
    float lse = m + logf(expf(c0 - m) + expf(c1 - m));
    lossc[idx] = lse - (pos ? c1 : c0);
    flags[idx] = (unsigned char)((pos ? 1 : 0) | (pos1 ? 2 : 0));
  }
  // reductions: every thread participates (EXEC all-ones for WMMA)
  float sL = block_sum(ll, lsum);
  float sLM = block_sum(llm, lsum);
  if (threadIdx.x == 0) {
    atomicAdd(&acc[0], sL);
    atomicAdd(&acc[2], sLM);
    atomicAdd(&numpos[b], cnt[0]);
    atomicAdd(&numposlm[b], cnt[1]);
  }
}

// Per-image top-k selection (hard negative mining) via 4-pass radix select on
// loss_c bits (all values > 0, so IEEE bits are monotone as u32), then an
// in-order tie scan so ties take lowest indices first (stable argsort match).
__global__ void k_select(const float* __restrict__ lossc,
                         const int* __restrict__ numpos,
                         unsigned char* __restrict__ flags) {
  __shared__ int hist[256];
  __shared__ int scan[256];
  __shared__ int sh_chosen, sh_above, sh_base;
  int b = blockIdx.x;
  int tid = threadIdx.x;
  int np = numpos[b];
  long long kk = (long long)NEGPOS * np;
  int k = (kk > (P_ - 1)) ? (P_ - 1) : (int)kk;
  if (k <= 0) return; // uniform exit; flags already hold only pos bits

  const float* lc = lossc + (size_t)b * P_;
  unsigned prefix = 0, fmask = 0;
  int rem = k;
  for (int shift = 24; shift >= 0; shift -= 8) {
    hist[tid] = 0;
    __syncthreads();
    for (int p = tid; p < P_; p += 256) {
      unsigned bits = __float_as_uint(lc[p]);
      if ((bits & fmask) == prefix) atomicAdd(&hist[(bits >> shift) & 255], 1);
    }
    __syncthreads();
    if (tid == 0) {
      int cum = 0, ch = 0;
      for (int h = 255; h >= 0; --h) {
        if (cum + hist[h] >= rem) { ch = h; break; }
        cum += hist[h];
      }
      sh_chosen = ch; sh_above = cum;
    }
    __syncthreads();
    prefix |= ((unsigned)sh_chosen) << shift;
    fmask |= 255u << shift;
    rem -= sh_above;
    __syncthreads();
  }
  unsigned T = prefix; // k-th largest value's bits; take 'rem' ties in order

  if (tid == 0) sh_base = 0;
  __syncthreads();
  for (int chunk = 0; chunk < PBLK; ++chunk) {
    int p = chunk * 256 + tid;
    int gt = 0, tie = 0;
    if (p < P_) {
      unsigned bits = __float_as_uint(lc[p]);
      gt = bits > T;
      tie = bits == T;
    }
    scan[tid] = tie;
    __syncthreads();
    for (int off = 1; off < 256; off <<= 1) {
      int v = scan[tid];
      int u = (tid >= off) ? scan[tid - off] : 0;
      __syncthreads();
      scan[tid] = v + u;
      __syncthreads();
    }
    int excl = scan[tid] - tie;
    int total = scan[255];
    int base = sh_base;
    bool neg = gt || (tie && (base + excl) < rem);
    if (p < P_ && neg) flags[(size_t)b * P_ + p] |= 4;
    __syncthreads();
    if (tid == 0) sh_base = base + total;
    __syncthreads();
  }
}

// Sum loss_c over pos|neg. grid (PBLK, B) x 256.
__global__ void k_negsum(const float* __restrict__ lossc,
                         const unsigned char* __restrict__ flags,
                         float* __restrict__ acc) {
  __shared__ float lsum[8];
  int b = blockIdx.y;
  int p = blockIdx.x * blockDim.x + threadIdx.x;
  float v = 0.f;
  if (p < P_) {
    size_t idx = (size_t)b * P_ + p;
    if (flags[idx] & 5u) v = lossc[idx];
  }
  float s = block_sum(v, lsum);
  if (threadIdx.x == 0) atomicAdd(&acc[1], s);
}

__global__ void k_final(const int* __restrict__ numpos,
                        const int* __restrict__ numposlm,
                        const float* __restrict__ acc, float* __restrict__ out) {
  if (threadIdx.x != 0 || blockIdx.x != 0) return;
  int sp = 0, sl = 0;
  for (int b = 0; b < B_; ++b) { sp += numpos[b]; sl += numposlm[b]; }
  float N = fmaxf((float)sp, 1.f);
  float N1 = fmaxf((float)sl, 1.f);
  out[0] = acc[0] / N;
  out[1] = acc[1] / N;
  out[2] = acc[2] / N1;
}

// ---------------------------------------------------------------------------
extern "C" void kernel_launch(void* const* d_in, const int* in_sizes, int n_in,
                              void* d_out, int out_size, void* d_ws, size_t ws_size,
                              hipStream_t stream) {
  const float* loc_data = (const float*)d_in[0];   // (B,P,4)
  const float* conf_data = (const float*)d_in[1];  // (B,P,2)
  const float* landm_data = (const float*)d_in[2]; // (B,P,10)
  const float* priors = (const float*)d_in[3];     // (P,4)
  const float* targets = (const float*)d_in[4];    // (B,O,15)
  float* out = (float*)d_out;

  const size_t np = (size_t)B_ * P_;
  char* ws = (char*)d_ws;
  float* bto = (float*)ws;                 ws += np * sizeof(float);
  int* bti = (int*)ws;                     ws += np * sizeof(int);
  float* lossc = (float*)ws;               ws += np * sizeof(float);
  unsigned char* flags = (unsigned char*)ws; ws += (np + 255) & ~(size_t)255;
  float* bpo = (float*)ws;                 ws += B_ * O_ * sizeof(float);
  int* bpi = (int*)ws;                     ws += B_ * O_ * sizeof(int);
  int* numpos = (int*)ws;                  ws += B_ * sizeof(int);
  int* numposlm = (int*)ws;                ws += B_ * sizeof(int);
  float* acc = (float*)ws;                 ws += 4 * sizeof(float);

  dim3 gridP(PBLK, B_);
  k_init<<<1, 64, 0, stream>>>(acc, numpos, numposlm);
  k_match<<<gridP, 256, 0, stream>>>(priors, targets, bto, bti);
  k_bestprior<<<dim3(O_, B_), 256, 0, stream>>>(priors, targets, bpo, bpi);
  k_override<<<B_, 64, 0, stream>>>(bpo, bpi, bto, bti);
  k_loss<<<gridP, 256, 0, stream>>>(loc_data, conf_data, landm_data, priors,
                                    targets, bto, bti, lossc, flags, numpos,
                                    numposlm, acc);
  k_select<<<B_, 256, 0, stream>>>(lossc, numpos, flags);
  k_negsum<<<gridP, 256, 0, stream>>>(lossc, flags, acc);
  k_final<<<1, 1, 0, stream>>>(numpos, numposlm, acc, out);
}